// MechanisticBackbone_76158360092832
// MI455X (gfx1250) — compile-verified
//
#include <hip/hip_runtime.h>

#define HD   128
#define BZ   2
#define NGn  20000
#define NPn  3000
#define EGG  200000
#define ETG  200000
#define EGP  200000
#define EPP  100000

typedef __attribute__((ext_vector_type(2))) float v2f;
typedef __attribute__((ext_vector_type(8))) float v8f;

// ---------------------------------------------------------------------------
// GEMM: Y[r, :] = act(X[r, :] @ W + bias), X:[rows,128], W:[128,128]
// One block = 256 threads = 8 waves; block computes a 16x128 row tile,
// wave w computes the 16x16 column tile at n0 = 16*w using
// V_WMMA_F32_16X16X4_F32 over 32 K-steps. A-tile staged in LDS.
// rows must be a multiple of 16 (40000 and 6000 both are).
// ---------------------------------------------------------------------------
__global__ __launch_bounds__(256) void gemm_h128(
    const float* __restrict__ X, const float* __restrict__ W,
    const float* __restrict__ bias, float* __restrict__ Y, int relu)
{
  __shared__ float sA[16][132];  // pad 132 -> conflict-free column reads
  const int tid  = threadIdx.x;
  const int wave = tid >> 5;
  const int lane = tid & 31;
  const size_t r0 = (size_t)blockIdx.x * 16;

  {  // cooperative A-tile load: 16 rows x 128 cols, 8 floats per thread
    int row = tid >> 4;
    int c0  = (tid & 15) * 8;
    const float* src = X + (r0 + row) * HD + c0;
#pragma unroll
    for (int j = 0; j < 8; ++j) sA[row][c0 + j] = src[j];
  }
  __syncthreads();

  const int n0 = wave * 16;
  const int m  = lane & 15;
  const int kh = (lane >> 4) * 2;  // 0 or 2: K-offset for upper lane half
  v8f c = {};
#pragma unroll
  for (int k0 = 0; k0 < HD; k0 += 4) {
    v2f a, b;
    a.x = sA[m][k0 + kh];
    a.y = sA[m][k0 + kh + 1];
    b.x = W[(size_t)(k0 + kh) * HD + n0 + m];
    b.y = W[(size_t)(k0 + kh + 1) * HD + n0 + m];
    c = __builtin_amdgcn_wmma_f32_16x16x4_f32(false, a, false, b,
                                              (short)0, c, false, false);
  }

  const float bv = bias ? bias[n0 + m] : 0.0f;
  const int rb = (lane >> 4) * 8;  // C/D: VGPR i -> row i (+8 for lanes 16..31)
#pragma unroll
  for (int i = 0; i < 8; ++i) {
    float v = c[i] + bv;
    if (relu) v = fmaxf(v, 0.0f);
    Y[(r0 + rb + i) * HD + n0 + m] = v;
  }
}

// ---------------------------------------------------------------------------
// Weighted edge scatter: Out[b, dst[e], :] += w(b,e) * T[b, src[e], :]
// one wave per (edge, batch); 4 floats per lane.
// ---------------------------------------------------------------------------
__global__ void scatter_msg(const float* __restrict__ T,
                            const int* __restrict__ srcI,
                            const int* __restrict__ dstI,
                            const float* __restrict__ ew, long ew_bstride,
                            float* __restrict__ Out, int E, int nsrc, int ndst)
{
  long wid = (long)blockIdx.x * (blockDim.x >> 5) + (threadIdx.x >> 5);
  int lane = threadIdx.x & 31;
  if (wid >= (long)E * BZ) return;
  int e = (int)(wid % E);
  int b = (int)(wid / E);
  float w = ew[ew_bstride * b + e];
  if (w == 0.0f) return;
  const float4 v = *(const float4*)(T + ((size_t)b * nsrc + srcI[e]) * HD + lane * 4);
  float* d = Out + ((size_t)b * ndst + dstI[e]) * HD + lane * 4;
  atomicAdd(d + 0, v.x * w);
  atomicAdd(d + 1, v.y * w);
  atomicAdd(d + 2, v.z * w);
  atomicAdd(d + 3, v.w * w);
}

// ---------------------------------------------------------------------------
// Edge attention scores: S[b,e] = sigmoid( dot(Q[b,dst,:],K[b,src,:]) / sqrt(H) * ew[e] )
// one wave per (edge, batch); wave-level shuffle reduction (wave32).
// ---------------------------------------------------------------------------
__global__ void edge_scores(const float* __restrict__ Q, const float* __restrict__ K,
                            const int* __restrict__ srcI, const int* __restrict__ dstI,
                            const float* __restrict__ ew, float* __restrict__ S,
                            int E, int nq, int nk)
{
  long wid = (long)blockIdx.x * (blockDim.x >> 5) + (threadIdx.x >> 5);
  int lane = threadIdx.x & 31;
  if (wid >= (long)E * BZ) return;
  int e = (int)(wid % E);
  int b = (int)(wid / E);
  const float4 qa = *(const float4*)(Q + ((size_t)b * nq + dstI[e]) * HD + lane * 4);
  const float4 kb = *(const float4*)(K + ((size_t)b * nk + srcI[e]) * HD + lane * 4);
  float r = qa.x * kb.x + qa.y * kb.y + qa.z * kb.z + qa.w * kb.w;
#pragma unroll
  for (int off = 16; off > 0; off >>= 1) r += __shfl_xor(r, off, 32);
  if (lane == 0) {
    float raw = r * 0.08838834764831845f;  // 1/sqrt(128)
    S[(size_t)b * E + e] = 1.0f / (1.0f + expf(-raw * ew[e]));
  }
}

// --------------------------- CSR-by-dst build ------------------------------
__global__ void zero_i32(int* p, int n) {
  int i = blockIdx.x * 256 + threadIdx.x;
  if (i < n) p[i] = 0;
}
__global__ void copy_i32(const int* __restrict__ a, int* __restrict__ b, int n) {
  int i = blockIdx.x * 256 + threadIdx.x;
  if (i < n) b[i] = a[i];
}
__global__ void csr_count(const int* __restrict__ dstI, int* cnt, int E) {
  int i = blockIdx.x * 256 + threadIdx.x;
  if (i < E) atomicAdd(&cnt[dstI[i]], 1);
}
__global__ void csr_fill(const int* __restrict__ dstI, int* cur, int* eid, int E) {
  int i = blockIdx.x * 256 + threadIdx.x;
  if (i < E) { int p = atomicAdd(&cur[dstI[i]], 1); eid[p] = i; }
}
// single-block exclusive scan (N <= a few 100k; here 20000 / 3000)
__global__ __launch_bounds__(1024) void ex_scan(const int* __restrict__ cnt,
                                                int* __restrict__ off, int N)
{
  __shared__ int sm[1024];
  __shared__ int carry;
  int tid = threadIdx.x;
  if (tid == 0) carry = 0;
  __syncthreads();
  for (int base = 0; base < N; base += 1024) {
    int i = base + tid;
    int x = (i < N) ? cnt[i] : 0;
    sm[tid] = x;
    __syncthreads();
    for (int d = 1; d < 1024; d <<= 1) {
      int t = (tid >= d) ? sm[tid - d] : 0;
      __syncthreads();
      sm[tid] += t;
      __syncthreads();
    }
    if (i < N) off[i] = carry + sm[tid] - x;
    int total = sm[1023];
    __syncthreads();
    if (tid == 0) carry += total;
    __syncthreads();
  }
  if (tid == 0) off[N] = carry;
}

// ---------------------------------------------------------------------------
// Top-k + threshold prune (matches lexsort((-s, dst)) stable ranking):
// keep edge e iff s>tau and #{e' same dst : s'>s or (s'==s and e'<e)} < topk.
// ---------------------------------------------------------------------------
__global__ void prune_topk(const float* __restrict__ S, const int* __restrict__ dstI,
                           const int* __restrict__ off, const int* __restrict__ eid,
                           float* __restrict__ Sel, int E, int topk, float tau)
{
  long g = (long)blockIdx.x * blockDim.x + threadIdx.x;
  if (g >= (long)E * BZ) return;
  int e = (int)(g % E);
  int b = (int)(g / E);
  const float* Sb = S + (size_t)b * E;
  float s = Sb[e];
  float out = 0.0f;
  if (s > tau) {
    int d = dstI[e];
    int lo = off[d], hi = off[d + 1];
    int better = 0;
    for (int j = lo; j < hi; ++j) {
      int e2 = eid[j];
      float s2 = Sb[e2];
      better += (int)((s2 > s) | ((s2 == s) & (e2 < e)));
    }
    if (better < topk) out = s;
  }
  Sel[(size_t)b * E + e] = out;
}

// ===========================================================================
extern "C" void kernel_launch(void* const* d_in, const int* in_sizes, int n_in,
                              void* d_out, int out_size, void* d_ws, size_t ws_size,
                              hipStream_t stream)
{
  const float* gene = (const float*)d_in[0];
  const float* path = (const float*)d_in[1];
  const float* gg_w = (const float*)d_in[2];
  const float* tg_w = (const float*)d_in[3];
  const float* gp_w = (const float*)d_in[4];
  const float* pp_w = (const float*)d_in[5];
  const int* gg_src = (const int*)d_in[6];
  const int* gg_dst = (const int*)d_in[7];
  const int* tg_src = (const int*)d_in[8];
  const int* tg_dst = (const int*)d_in[9];
  const int* gp_src = (const int*)d_in[10];
  const int* gp_dst = (const int*)d_in[11];
  const int* pp_src = (const int*)d_in[12];
  const int* pp_dst = (const int*)d_in[13];
  const float* mgg_w1 = (const float*)d_in[14]; const float* mgg_b1 = (const float*)d_in[15];
  const float* mgg_w2 = (const float*)d_in[16]; const float* mgg_b2 = (const float*)d_in[17];
  const float* mgp_w1 = (const float*)d_in[18]; const float* mgp_b1 = (const float*)d_in[19];
  const float* mgp_w2 = (const float*)d_in[20]; const float* mgp_b2 = (const float*)d_in[21];
  const float* mpp_w1 = (const float*)d_in[22]; const float* mpp_b1 = (const float*)d_in[23];
  const float* mpp_w2 = (const float*)d_in[24]; const float* mpp_b2 = (const float*)d_in[25];
  const float* mpg_w1 = (const float*)d_in[26]; const float* mpg_b1 = (const float*)d_in[27];
  const float* mpg_w2 = (const float*)d_in[28]; const float* mpg_b2 = (const float*)d_in[29];
  const float* w_gg = (const float*)d_in[30];
  const float* w_pp = (const float*)d_in[31];
  const float* w_pg = (const float*)d_in[32];
  const float* tg_q = (const float*)d_in[33];
  const float* tg_k = (const float*)d_in[34];
  const float* tg_vt = (const float*)d_in[35];
  const float* tg_vg = (const float*)d_in[36];
  const float* gp_q = (const float*)d_in[37];
  const float* gp_k = (const float*)d_in[38];
  const float* gp_v = (const float*)d_in[39];

  float* out_final = (float*)d_out;                       // [B,NG,H]
  float* out_h3    = out_final + (size_t)BZ * NGn * HD;   // [B,NP,H]

  // ---- workspace carve-out ----
  char* wptr = (char*)d_ws;
  auto carve = [&](size_t bytes) { char* p = wptr; wptr += (bytes + 255) & ~(size_t)255; return p; };
  const size_t GSZ = sizeof(float) * (size_t)BZ * NGn * HD;
  const size_t PSZ = sizeof(float) * (size_t)BZ * NPn * HD;
  float* G0 = (float*)carve(GSZ);
  float* G1 = (float*)carve(GSZ);
  float* G2 = (float*)carve(GSZ);
  float* P0 = (float*)carve(PSZ);
  float* P1 = (float*)carve(PSZ);
  float* P2 = (float*)carve(PSZ);
  float* S_tg   = (float*)carve(sizeof(float) * (size_t)BZ * ETG);
  float* SEL_tg = (float*)carve(sizeof(float) * (size_t)BZ * ETG);
  float* S_gp   = (float*)carve(sizeof(float) * (size_t)BZ * EGP);
  float* SEL_gp = (float*)carve(sizeof(float) * (size_t)BZ * EGP);
  int* tg_off = (int*)carve(sizeof(int) * (NGn + 1));
  int* tg_cur = (int*)carve(sizeof(int) * NGn);
  int* tg_eid = (int*)carve(sizeof(int) * ETG);
  int* gp_off = (int*)carve(sizeof(int) * (NPn + 1));
  int* gp_cur = (int*)carve(sizeof(int) * NPn);
  int* gp_eid = (int*)carve(sizeof(int) * EGP);
  int* cnt    = (int*)carve(sizeof(int) * NGn);

  const int RG = BZ * NGn;   // 40000 rows (mult of 16)
  const int RP = BZ * NPn;   // 6000 rows (mult of 16)
  auto GEMM = [&](const float* X, const float* W, const float* Bv, float* Y,
                  int rows, int relu) {
    gemm_h128<<<rows / 16, 256, 0, stream>>>(X, W, Bv, Y, relu);
  };
  auto wave_blocks = [](long E) { return (int)((E * BZ + 7) / 8); };   // 8 waves/block
  auto thr_blocks  = [](long E) { return (int)((E * BZ + 255) / 256); };

  // ---- CSR by dst (batch-independent) ----
  zero_i32<<<(NGn + 255) / 256, 256, 0, stream>>>(cnt, NGn);
  csr_count<<<(ETG + 255) / 256, 256, 0, stream>>>(tg_dst, cnt, ETG);
  ex_scan<<<1, 1024, 0, stream>>>(cnt, tg_off, NGn);
  copy_i32<<<(NGn + 255) / 256, 256, 0, stream>>>(tg_off, tg_cur, NGn);
  csr_fill<<<(ETG + 255) / 256, 256, 0, stream>>>(tg_dst, tg_cur, tg_eid, ETG);

  zero_i32<<<(NPn + 255) / 256, 256, 0, stream>>>(cnt, NPn);
  csr_count<<<(EGP + 255) / 256, 256, 0, stream>>>(gp_dst, cnt, EGP);
  ex_scan<<<1, 1024, 0, stream>>>(cnt, gp_off, NPn);
  copy_i32<<<(NPn + 255) / 256, 256, 0, stream>>>(gp_off, gp_cur, NPn);
  csr_fill<<<(EGP + 255) / 256, 256, 0, stream>>>(gp_dst, gp_cur, gp_eid, EGP);

  // ---- Layer 1: gene-gene ----
  GEMM(gene, w_gg, nullptr, G0, RG, 0);                 // gg_t = gene @ w_gg
  GEMM(gene, mgg_w1, mgg_b1, G1, RG, 1);                // mid  = relu(gene@w1+b1)
  GEMM(G1, mgg_w2, mgg_b2, G2, RG, 0);                  // h1   = mid@w2+b2
  scatter_msg<<<wave_blocks(EGG), 256, 0, stream>>>(G0, gg_src, gg_dst, gg_w, 0,
                                                    G2, EGG, NGn, NGn);  // h1 += msgs

  // ---- Layer 2: TF->gene attention ----
  GEMM(G2, tg_q, nullptr, G0, RG, 0);                   // Q (by dst)
  GEMM(G2, tg_k, nullptr, G1, RG, 0);                   // K (by src)
  edge_scores<<<wave_blocks(ETG), 256, 0, stream>>>(G0, G1, tg_src, tg_dst, tg_w,
                                                    S_tg, ETG, NGn, NGn);
  prune_topk<<<thr_blocks(ETG), 256, 0, stream>>>(S_tg, tg_dst, tg_off, tg_eid,
                                                  SEL_tg, ETG, 16, 0.5f);
  GEMM(G2, tg_vt, nullptr, G0, RG, 0);                  // v_tf   (Q dead)
  GEMM(G2, tg_vg, nullptr, G1, RG, 0);                  // h2 base = v_gene (K dead)
  scatter_msg<<<wave_blocks(ETG), 256, 0, stream>>>(G0, tg_src, tg_dst, SEL_tg, (long)ETG,
                                                    G1, ETG, NGn, NGn);  // h2 = G1

  // ---- Layer 3: gene->pathway attention ----
  GEMM(G1, gp_k, nullptr, G0, RG, 0);                   // k_g (h1 dead -> G2 free)
  GEMM(G1, gp_v, nullptr, G2, RG, 0);                   // v_g
  GEMM(path, gp_q, nullptr, P0, RP, 0);                 // q_p
  GEMM(path, mgp_w1, mgp_b1, P1, RP, 1);
  GEMM(P1, mgp_w2, mgp_b2, P2, RP, 0);                  // bar_h3 base
  edge_scores<<<wave_blocks(EGP), 256, 0, stream>>>(P0, G0, gp_src, gp_dst, gp_w,
                                                    S_gp, EGP, NPn, NGn);
  prune_topk<<<thr_blocks(EGP), 256, 0, stream>>>(S_gp, gp_dst, gp_off, gp_eid,
                                                  SEL_gp, EGP, 32, 0.5f);
  scatter_msg<<<wave_blocks(EGP), 256, 0, stream>>>(G2, gp_src, gp_dst, SEL_gp, (long)EGP,
                                                    P2, EGP, NGn, NPn);  // bar_h3 = P2

  // ---- Layer 4: pathway-pathway ----
  GEMM(P2, w_pp, nullptr, P0, RP, 0);                   // pp_t
  GEMM(P2, mpp_w1, mpp_b1, P1, RP, 1);
  GEMM(P1, mpp_w2, mpp_b2, out_h3, RP, 0);              // h3 base -> output
  scatter_msg<<<wave_blocks(EPP), 256, 0, stream>>>(P0, pp_src, pp_dst, pp_w, 0,
                                                    out_h3, EPP, NPn, NPn);

  // ---- Layer 5: pathway->gene feedback ----
  GEMM(out_h3, w_pg, nullptr, P0, RP, 0);               // pg_t = h3 @ w_pg
  GEMM(G1, mpg_w1, mpg_b1, G0, RG, 1);                  // mid = relu(h2@w1+b1)
  GEMM(G0, mpg_w2, mpg_b2, out_final, RG, 0);           // final base
  scatter_msg<<<wave_blocks(EGP), 256, 0, stream>>>(P0, gp_dst, gp_src, gp_w, 0,
                                                    out_final, EGP, NPn, NGn);

  (void)in_sizes; (void)n_in; (void)out_size; (void)ws_size;
}